// NCutsLoss_66133906424400
// MI455X (gfx1250) — compile-verified
//
#include <hip/hip_runtime.h>

#define B_      4
#define K_      4
#define H_      224
#define W_      224
#define WIN2    81
#define CH2     108     // per-pixel weight stride in LDS: 9 rows x 12 taps (cols 9..11 zero)
#define NCHUNK  27      // 27 chunks of K=4 taps, 3 per window row
#define HALFW   112     // pixels per workgroup (half a row)
#define SEGW    128     // padded seg-window row stride in LDS (needs >= 123)
#define PLANE   (9*SEGW)  // 1152 floats per k-plane
#define NWAVE   7
#define NTHREADS 224
#define NWG     (B_*H_*2)   // 1792

typedef __attribute__((ext_vector_type(2))) float v2f;
typedef __attribute__((ext_vector_type(4))) float v4f;
typedef __attribute__((ext_vector_type(8))) float v8f;

__device__ __forceinline__ float pick8(v8f v, int r) {
    float d =            v[0];
    d = (r == 1) ? v[1] : d;
    d = (r == 2) ? v[2] : d;
    d = (r == 3) ? v[3] : d;
    d = (r == 4) ? v[4] : d;
    d = (r == 5) ? v[5] : d;
    d = (r == 6) ? v[6] : d;
    d = (r == 7) ? v[7] : d;
    return d;
}

__device__ __forceinline__ float wave_sum(float x) {
    #pragma unroll
    for (int off = 16; off; off >>= 1) x += __shfl_xor(x, off, 32);
    return x;
}

__global__ __launch_bounds__(NTHREADS)
void ncuts_partial(const float* __restrict__ seg, const float* __restrict__ wgt,
                   float* __restrict__ part)
{
    __shared__ float w_lds[HALFW * CH2];       // 48384 B: weight, rows padded 9->12 taps
    __shared__ float s_lds[K_ * PLANE];        // 18432 B: zero-padded seg window rows
    __shared__ float sw_lds[HALFW];            // per-pixel sum of weights
    __shared__ float red[NWAVE * 8];           // per-wave partials [wave][k][A/V]

    const int tid = threadIdx.x;
    const int wg  = blockIdx.x;
    const int b   = wg / (H_ * 2);
    const int rem = wg - b * (H_ * 2);
    const int h   = rem >> 1;
    const int w0  = (rem & 1) * HALFW;

    // ---- phase 0: zero seg staging + weight pad taps (cols 9..11 of each row) ----
    for (int e = tid; e < K_ * PLANE; e += NTHREADS) s_lds[e] = 0.f;
    for (int e = tid; e < HALFW * 27; e += NTHREADS) {
        int p = e / 27;
        int i = e - 27 * p;
        int r = i / 3;
        w_lds[p * CH2 + r * 12 + 9 + (i - 3 * r)] = 0.f;
    }
    __syncthreads();

    // ---- phase 1: stream this row-half's weights (dominant traffic) as b128 ----
    {
        const v4f* gw4 = (const v4f*)(wgt + ((size_t)(b * H_ + h) * W_ + w0) * WIN2);
        for (int e4 = tid; e4 < (HALFW * WIN2) / 4; e4 += NTHREADS) {   // 2268 vec4s
            v4f v = gw4[e4];
            int idx = 4 * e4;
            int p   = idx / 81;
            int o   = idx - 81 * p;
            int r   = o / 9;
            int col = o - 9 * r;
            int base = p * CH2 + r * 12 + col;
            float vv[4] = {v.x, v.y, v.z, v.w};
            #pragma unroll
            for (int j = 0; j < 4; ++j) {
                w_lds[base] = vv[j];
                ++col; ++base;
                if (col == 9) { col = 0; base += 3; }  // skip pad; 9*12=108 => pixel wrap is free
            }
        }
    }
    // seg window rows h-5..h+3, cols w0-5..w0+114, zero-padded borders (= jnp.pad)
    for (int e = tid; e < K_ * 9 * 120; e += NTHREADS) {
        int k  = e / 1080;
        int r1 = e - 1080 * k;
        int r  = r1 / 120;
        int c  = r1 - 120 * r;
        int hh = h + r - 5;
        int ww = w0 + c - 5;
        if (hh >= 0 && hh < H_ && ww >= 0 && ww < W_)
            s_lds[(k * 9 + r) * SEGW + c] =
                seg[(((size_t)b * K_ + k) * H_ + hh) * W_ + ww];
    }
    __syncthreads();

    // ---- phase 2: per-pixel window weight sums (pad taps are zero, fold them in) ----
    for (int p = tid; p < HALFW; p += NTHREADS) {
        const v4f* row = (const v4f*)(w_lds + p * CH2);  // 432B row stride: 16B aligned
        float s = 0.f;
        #pragma unroll
        for (int j = 0; j < CH2 / 4; ++j) {
            v4f v = row[j];
            s += v.x + v.y + v.z + v.w;
        }
        sw_lds[p] = s;
    }
    __syncthreads();

    // ---- phase 3: WMMA over 27 chunks; wave t owns pixels 16t..16t+15 ----
    // All chunk/plane offsets are compile-time immediates off two per-lane bases.
    const int wave = tid >> 5;
    const int lane = tid & 31;
    const int ln   = lane & 15;
    const int hi   = lane >> 4;          // A: lanes 0-15 hold K={0,1}, lanes 16-31 K={2,3}
    const int t    = wave;
    const int p    = t * 16 + ln;        // A row M / B col N pixel
    const float* wpA = w_lds + p * CH2 + 2 * hi;
    const float* sb  = s_lds + p + 2 * hi;

    v8f acc0 = {}, acc1 = {}, acc2 = {}, acc3 = {};
    #pragma unroll
    for (int c = 0; c < NCHUNK; ++c) {
        const int r   = c / 3;               // compile-time after unroll
        const int cb  = 4 * (c - 3 * r);
        const int ao  = 12 * r + cb;         // A tap offset (even -> b64-aligned)
        const int off = r * SEGW + cb;       // B offset within a k-plane
        v2f a;
        a.x = wpA[ao];
        a.y = wpA[ao + 1];
        v2f bv;
        bv.x = sb[off];              bv.y = sb[off + 1];
        acc0 = __builtin_amdgcn_wmma_f32_16x16x4_f32(false, a, false, bv, (short)0, acc0, false, false);
        bv.x = sb[off + PLANE];      bv.y = sb[off + PLANE + 1];
        acc1 = __builtin_amdgcn_wmma_f32_16x16x4_f32(false, a, false, bv, (short)0, acc1, false, false);
        bv.x = sb[off + 2*PLANE];    bv.y = sb[off + 2*PLANE + 1];
        acc2 = __builtin_amdgcn_wmma_f32_16x16x4_f32(false, a, false, bv, (short)0, acc2, false, false);
        bv.x = sb[off + 3*PLANE];    bv.y = sb[off + 3*PLANE + 1];
        acc3 = __builtin_amdgcn_wmma_f32_16x16x4_f32(false, a, false, bv, (short)0, acc3, false, false);
    }

    // ---- phase 4: extract D diagonal, fold with seg & sumW, wave-reduce ----
    // D layout: vgpr r, lanes 0-15 -> (M=r, N=lane); lanes 16-31 -> (M=r+8, N=lane-16).
    // Diagonal: lanes 0-7 (i=lane, r=lane) and lanes 24-31 (i=lane-16, r=lane-24).
    const bool lo8  = lane < 8;
    const bool hi8  = lane >= 24;
    const bool act  = lo8 || hi8;
    const int  rr   = lo8 ? lane : (hi8 ? lane - 24 : 0);
    const int  ii   = lo8 ? lane : (hi8 ? lane - 16 : 0);
    const int  px   = t * 16 + ii;
    const float swv = sw_lds[px];

    #pragma unroll
    for (int k = 0; k < K_; ++k) {
        v8f acc = (k == 0) ? acc0 : (k == 1) ? acc1 : (k == 2) ? acc2 : acc3;
        float d = pick8(acc, rr);
        d = act ? d : 0.f;
        // seg value at (h, w0+px): window row index 5, col px+5
        float segv = s_lds[(k * 9 + 5) * SEGW + px + 5];
        float cA = d * segv;
        float cV = act ? swv * segv : 0.f;
        cA = wave_sum(cA);
        cV = wave_sum(cV);
        if (lane == 0) {
            red[wave * 8 + k * 2 + 0] = cA;
            red[wave * 8 + k * 2 + 1] = cV;
        }
    }
    __syncthreads();

    // ---- phase 5: fixed-order block reduce -> per-workgroup partials ----
    if (tid < 8) {
        int k = tid & 3, which = tid >> 2;
        float s = 0.f;
        #pragma unroll
        for (int w = 0; w < NWAVE; ++w) s += red[w * 8 + k * 2 + which];
        part[((size_t)wg * K_ + k) * 2 + which] = s;
    }
}

__global__ __launch_bounds__(32)
void ncuts_finalize(const float* __restrict__ part, float* __restrict__ out)
{
    const int lane = threadIdx.x;
    float r = 0.f;
    if (lane < 16) {
        int b = lane >> 2, k = lane & 3;
        float A = 0.f, V = 0.f;
        for (int i = 0; i < H_ * 2; ++i) {          // fixed order: deterministic
            const float* pp = part + (((size_t)(b * H_ * 2 + i)) * K_ + k) * 2;
            A += pp[0];
            V += pp[1];
        }
        r = A / V;
    }
    #pragma unroll
    for (int off = 16; off; off >>= 1) r += __shfl_xor(r, off, 32);
    if (lane == 0) out[0] = (float)(B_ * K_) - r;   // sum_b (K - assoc_b) = 16 - sum(A/V)
}

extern "C" void kernel_launch(void* const* d_in, const int* in_sizes, int n_in,
                              void* d_out, int out_size, void* d_ws, size_t ws_size,
                              hipStream_t stream)
{
    (void)in_sizes; (void)n_in; (void)out_size; (void)ws_size;
    const float* seg = (const float*)d_in[0];   // [4,4,224,224] f32
    const float* wgt = (const float*)d_in[1];   // [4,1,224,224,9,9] f32
    float* part = (float*)d_ws;                 // [1792][4][2] partials (fully rewritten each call)
    float* out  = (float*)d_out;

    ncuts_partial<<<NWG, NTHREADS, 0, stream>>>(seg, wgt, part);
    ncuts_finalize<<<1, 32, 0, stream>>>(part, out);
}